// KnnExpansion_30829275251161
// MI455X (gfx1250) — compile-verified
//
#include <hip/hip_runtime.h>

// ---------------------------------------------------------------------------
// KnnExpansion on MI455X (gfx1250):
//   out[f,n] = sum_k alpha[i[n,k], f] * exp(-0.5*d[n,k]/sigma^2)
// Strategy: one wave32 per query. TDM gather-mode descriptors asynchronously
// pull the 32 indexed alpha rows (256B each) into LDS (double-buffered across
// queries, tracked with TENSORcnt), while the wave computes the previous
// query's weighted row-sum with VALU FMAs. alpha (16MB) lives in L2 (192MB),
// so the TDM gather is an L2->LDS pipeline.
// ---------------------------------------------------------------------------

#define KNN_K     32      // neighbors per query
#define KNN_F     64      // alpha channels
#define WAVES_PB  4       // waves per block
#define QPW       16      // queries processed per wave
#define ROW_BYTES (KNN_F * 4)          // 256 B per gathered row
#define BUF_FLOATS (KNN_K * KNN_F)     // 2048 floats = 8 KB per buffer

typedef __attribute__((ext_vector_type(4))) unsigned int v4u;
typedef __attribute__((ext_vector_type(8))) int          v8i;
typedef __attribute__((ext_vector_type(4))) int          v4i;

// The TDM writes LDS behind the compiler's back. Escape the LDS array's
// address into an opaque asm with a memory clobber so the optimizer must
// assume its contents are (re)written externally — otherwise the
// "never-stored global -> loads are undef" fold deletes the whole compute.
__device__ __forceinline__ void tdm_lds_publish(const float* p) {
    asm volatile("" :: "v"(p) : "memory");
}

// Issue one TDM gather of 16 indexed rows (16-bit index mode) into LDS.
// ldsOff: byte offset of destination within workgroup LDS allocation.
// ga:     global byte address of alpha base (row 0).
// ip:     pointer to 16 consecutive int32 row indices (wave-uniform).
__device__ __forceinline__ void tdm_gather16(unsigned ldsOff,
                                             unsigned long long ga,
                                             const int* __restrict__ ip) {
    v4u g0;
    g0[0] = 0x80000001u;                 // gather_mode=1, 16-bit idx, count=1
    g0[1] = ldsOff;                      // lds_addr
    g0[2] = (unsigned)ga;                // global_addr[31:0]
    g0[3] = ((unsigned)(ga >> 32) & 0x01FFFFFFu) | 0x80000000u; // addr[56:32], type=2

    v8i g1;
    g1[0] = 0x00020000;                  // data_size=2 (4B), wg_mask=0
    g1[1] = (KNN_F & 0xFFFF) << 16;      // tensor_dim0 = 64 elements/row
    g1[2] = 0;                           // tensor_dim0[31:16]=0, tensor_dim1[15:0]=0
    g1[3] = 1 | (KNN_F << 16);           // tensor_dim1[31:16]=1 (65536 rows), tile_dim0=64
    g1[4] = 16;                          // tile_dim1 = 16 valid indices
    g1[5] = KNN_F;                       // tensor_dim0_stride = 64 elements
    g1[6] = 0;
    g1[7] = 0;

    v4i g2, g3;
#pragma unroll
    for (int t = 0; t < 4; ++t)
        g2[t] = (ip[2 * t] & 0xFFFF) | (ip[2 * t + 1] << 16);
#pragma unroll
    for (int t = 0; t < 4; ++t)
        g3[t] = (ip[8 + 2 * t] & 0xFFFF) | (ip[8 + 2 * t + 1] << 16);

#if defined(__clang_major__) && (__clang_major__ >= 23)
    v8i gx = (v8i)0;
    __builtin_amdgcn_tensor_load_to_lds(g0, g1, g2, g3, gx, 0);
#else
    __builtin_amdgcn_tensor_load_to_lds(g0, g1, g2, g3, 0);
#endif
}

// Issue both 16-row gathers for one query's 32 neighbors.
__device__ __forceinline__ void issue_query(const int* __restrict__ ip,
                                            unsigned ldsBase,
                                            unsigned long long ga) {
    tdm_gather16(ldsBase, ga, ip);
    tdm_gather16(ldsBase + 16u * ROW_BYTES, ga, ip + 16);
}

__global__ __launch_bounds__(WAVES_PB * 32)
void KnnExpansion_30829275251161_kernel(const float* __restrict__ d,
                                        const int* __restrict__ idx,
                                        const float* __restrict__ alpha,
                                        const float* __restrict__ sigma,
                                        float* __restrict__ out,
                                        int Nq) {
    // 4 waves x 2 buffers x 8KB = 64 KB LDS
    __shared__ float lds[WAVES_PB * 2 * BUF_FLOATS];

    const int lane = threadIdx.x & 31;
    const int wid  = __builtin_amdgcn_readfirstlane((int)(threadIdx.x >> 5));
    const int baseQ = (blockIdx.x * WAVES_PB + wid) * QPW;
    if (baseQ >= Nq) return;

    // Visible store so the array is not a "never-written global" (the TDM
    // immediately overwrites it; the value is irrelevant).
    lds[wid * 2 * BUF_FLOATS + lane] = 0.0f;

    const float s  = sigma[0];
    const float cc = -0.5f / (s * s);
    const unsigned long long ga = (unsigned long long)(uintptr_t)alpha;
    const int* __restrict__ ipBase = idx + (size_t)baseQ * KNN_K;

    const unsigned waveLdsByte = (unsigned)(wid * 2 * BUF_FLOATS * 4);

    // Prologue: gather query baseQ into buffer 0.
    issue_query(ipBase, waveLdsByte, ga);

#pragma unroll 2
    for (int j = 0; j < QPW; ++j) {
        const int n   = baseQ + j;
        const int sel = j & 1;

        if (j + 1 < QPW) {
            // WAR guard: prior iteration's LDS reads of buffer sel^1 must
            // have completed before TDM overwrites it.
            asm volatile("s_wait_dscnt 0" ::: "memory");
            issue_query(ipBase + (j + 1) * KNN_K,
                        waveLdsByte + (unsigned)((sel ^ 1) * BUF_FLOATS * 4), ga);
            // 2 ops just issued may remain outstanding; the 2 ops for buffer
            // `sel` (issued earlier, in-order) must be complete.
            __builtin_amdgcn_s_wait_tensorcnt(2);
        } else {
            __builtin_amdgcn_s_wait_tensorcnt(0);
        }

        const float* __restrict__ buf = &lds[(wid * 2 + sel) * BUF_FLOATS];
        // TDM results are now in LDS: force the compiler to treat the buffer
        // contents as externally written, and keep loads below the wait.
        tdm_lds_publish(buf);

        // Gaussian weights: lane l computes w for neighbor k=l (coalesced load).
        const float dv = d[(size_t)n * KNN_K + lane];
        const float wv = __expf(dv * cc);

        float accx = 0.0f, accy = 0.0f;
#pragma unroll
        for (int k = 0; k < KNN_K; ++k) {
            const float wk = __int_as_float(
                __builtin_amdgcn_readlane(__float_as_int(wv), k));
            // Row k: 256B, lanes read float2 -> ds_load_b64, conflict-free.
            const float2 a = *(const float2*)&buf[k * KNN_F + lane * 2];
            accx = fmaf(wk, a.x, accx);
            accy = fmaf(wk, a.y, accy);
        }

        // out is [F, N]; consecutive j in a wave -> consecutive n, L2 write-combines.
        out[(size_t)(2 * lane)     * Nq + n] = accx;
        out[(size_t)(2 * lane + 1) * Nq + n] = accy;
    }
}

extern "C" void kernel_launch(void* const* d_in, const int* in_sizes, int n_in,
                              void* d_out, int out_size, void* d_ws, size_t ws_size,
                              hipStream_t stream) {
    const float* d     = (const float*)d_in[0];
    const int*   idx   = (const int*)d_in[1];
    const float* alpha = (const float*)d_in[2];
    const float* sigma = (const float*)d_in[3];
    float*       out   = (float*)d_out;

    const int N = in_sizes[0] / KNN_K;           // 131072
    const int queriesPerBlock = WAVES_PB * QPW;  // 64
    const int blocks = (N + queriesPerBlock - 1) / queriesPerBlock;

    KnnExpansion_30829275251161_kernel<<<blocks, WAVES_PB * 32, 0, stream>>>(
        d, idx, alpha, sigma, out, N);
}